// TropEmbed_8091718386440
// MI455X (gfx1250) — compile-verified
//
#include <hip/hip_runtime.h>

// ---------------------------------------------------------------------------
// Tropical "embedding" distance:  out[b,u] = max_d(x[b,d]+w[u,d])
//                                          - min_d(x[b,d]+w[u,d]) + bias[u]
// B=8192, U=64, D=512, all fp32.  VALU-bound (max-plus semiring: WMMA N/A).
// CDNA5 paths used: wave32, async global->LDS copies (ASYNCcnt), big LDS,
// v_max3/v_min3 + v_pk_add_f32 friendly inner loop.
// ---------------------------------------------------------------------------

#define AS1 __attribute__((address_space(1)))
#define AS3 __attribute__((address_space(3)))

typedef float f32x2 __attribute__((ext_vector_type(2)));
typedef float f32x4 __attribute__((ext_vector_type(4)));
typedef int   v4i   __attribute__((ext_vector_type(4)));

constexpr int Bn = 8192;
constexpr int Un = 64;
constexpr int Dn = 512;

constexpr int UB = 32;          // u columns per block
constexpr int BB = 64;          // b rows per block
constexpr int DC = 128;         // d chunk streamed through LDS
constexpr int NC = Dn / DC;     // 4 chunks
constexpr int WS = 516;         // w LDS row stride (floats): 512 + 4 pad (16B aligned, conflict-free)
constexpr int XS = 132;         // x LDS row stride (floats): 128 + 4 pad

#if __has_builtin(__builtin_amdgcn_global_load_async_to_lds_b128)
#define HAVE_ASYNC 1
#else
#define HAVE_ASYNC 0
#endif

#if __has_builtin(__builtin_amdgcn_s_wait_asynccnt)
#define WAIT_ASYNC(n) __builtin_amdgcn_s_wait_asynccnt(n)
#else
#define WAIT_ASYNC(n) asm volatile("s_wait_asynccnt %0" ::"i"(n) : "memory")
#endif

__device__ __forceinline__ float fmax3(float a, float b, float c) {
  return __builtin_fmaxf(__builtin_fmaxf(a, b), c);   // -> v_max3_num_f32
}
__device__ __forceinline__ float fmin3(float a, float b, float c) {
  return __builtin_fminf(__builtin_fminf(a, b), c);   // -> v_min3_num_f32
}

// 16-byte global -> LDS copy: async DMA when the toolchain exposes it,
// otherwise a plain load+ds_store pair (still compiles/works everywhere).
__device__ __forceinline__ void copy16(const float* g, float* l) {
#if HAVE_ASYNC
  __builtin_amdgcn_global_load_async_to_lds_b128((AS1 v4i*)g,
                                                 (AS3 v4i*)l,
                                                 /*offset=*/0, /*cpol=*/0);
#else
  *(f32x4*)l = *(const f32x4*)g;
#endif
}

__global__ __launch_bounds__(256, 1) void trop_embed_kernel(
    const float* __restrict__ x, const float* __restrict__ w,
    const float* __restrict__ bias, float* __restrict__ out) {
  extern __shared__ __align__(16) float smem[];
  float* ws  = smem;              // [UB][WS]   66,048 B
  float* xs0 = smem + UB * WS;    // [BB][XS]   33,792 B
  float* xs1 = xs0 + BB * XS;     // [BB][XS]   33,792 B

  const int tid = threadIdx.x;
  const int tx  = tid & 7;        // u-group: covers u = tx + 8*i, i<4
  const int ty  = tid >> 3;       // b-group: covers b = 2*ty + j,  j<2
  const int u0  = blockIdx.x * UB;
  const int b0  = blockIdx.y * BB;

  // ---- stage W tile (UB x Dn) into LDS: 4096 float4 / 256 thr = 16 each ----
#pragma unroll
  for (int k = 0; k < 16; ++k) {
    const int f   = tid + k * 256;
    const int row = f >> 7;               // f / 128
    const int col = (f & 127) << 2;       // 16B aligned
    copy16(w + (size_t)(u0 + row) * Dn + col, ws + row * WS + col);
  }
  // ---- prefetch x chunk 0: 2048 float4 / 256 thr = 8 each ----
#pragma unroll
  for (int k = 0; k < 8; ++k) {
    const int f   = tid + k * 256;
    const int row = f >> 5;               // f / 32
    const int col = (f & 31) << 2;
    copy16(x + (size_t)(b0 + row) * Dn + col, xs0 + row * XS + col);
  }

  float mx[2][4], mn[2][4];
#pragma unroll
  for (int j = 0; j < 2; ++j)
#pragma unroll
    for (int i = 0; i < 4; ++i) {
      mx[j][i] = -__builtin_inff();
      mn[j][i] = __builtin_inff();
    }

  for (int c = 0; c < NC; ++c) {
    float* xb = (c & 1) ? xs1 : xs0;
    if (c + 1 < NC) {
      float* nb = (c & 1) ? xs0 : xs1;    // prefetch next chunk
#pragma unroll
      for (int k = 0; k < 8; ++k) {
        const int f   = tid + k * 256;
        const int row = f >> 5;
        const int col = (f & 31) << 2;
        copy16(x + (size_t)(b0 + row) * Dn + (c + 1) * DC + col,
               nb + row * XS + col);
      }
#if HAVE_ASYNC
      WAIT_ASYNC(8);    // 8 newly-issued ops may stay in flight; older done
#endif
    } else {
#if HAVE_ASYNC
      WAIT_ASYNC(0);
#endif
    }
    __syncthreads();

    const float* xr0 = xb + (2 * ty) * XS;
    const float* xr1 = xr0 + XS;
    const float* wp[4] = {ws + (tx + 0) * WS + c * DC,
                          ws + (tx + 8) * WS + c * DC,
                          ws + (tx + 16) * WS + c * DC,
                          ws + (tx + 24) * WS + c * DC};

#pragma unroll 4
    for (int d2 = 0; d2 < DC; d2 += 2) {
      const f32x2 a0 = *(const f32x2*)(xr0 + d2);   // ds_load_b64
      const f32x2 a1 = *(const f32x2*)(xr1 + d2);
#pragma unroll
      for (int i = 0; i < 4; ++i) {
        const f32x2 v  = *(const f32x2*)(wp[i] + d2);
        const f32x2 s0 = a0 + v;                    // -> v_pk_add_f32
        const f32x2 s1 = a1 + v;
        mx[0][i] = fmax3(mx[0][i], s0.x, s0.y);
        mn[0][i] = fmin3(mn[0][i], s0.x, s0.y);
        mx[1][i] = fmax3(mx[1][i], s1.x, s1.y);
        mn[1][i] = fmin3(mn[1][i], s1.x, s1.y);
      }
    }
    __syncthreads();   // before next iteration's prefetch reuses a buffer
  }

  // ---- epilogue: out[b,u] = mx - mn + bias[u] ----
#pragma unroll
  for (int j = 0; j < 2; ++j) {
    const int b = b0 + 2 * ty + j;
#pragma unroll
    for (int i = 0; i < 4; ++i) {
      const int u = u0 + tx + 8 * i;
      out[(size_t)b * Un + u] = mx[j][i] - mn[j][i] + bias[u];
    }
  }
}

extern "C" void kernel_launch(void* const* d_in, const int* in_sizes, int n_in,
                              void* d_out, int out_size, void* d_ws,
                              size_t ws_size, hipStream_t stream) {
  (void)in_sizes; (void)n_in; (void)out_size; (void)d_ws; (void)ws_size;
  const float* x    = (const float*)d_in[0];
  const float* w    = (const float*)d_in[1];
  const float* bias = (const float*)d_in[2];
  float* out        = (float*)d_out;

  const size_t shmem = (size_t)(UB * WS + 2 * BB * XS) * sizeof(float); // 133,632 B
  dim3 grid(Un / UB, Bn / BB);   // (2, 128) = 256 workgroups
  dim3 block(256);               // 8 wave32
  trop_embed_kernel<<<grid, block, shmem, stream>>>(x, w, bias, out);
}